// EMDLoss_39771397161590
// MI455X (gfx1250) — compile-verified
//
#include <hip/hip_runtime.h>
#include <math.h>

typedef __attribute__((ext_vector_type(2))) float v2f;
typedef __attribute__((ext_vector_type(8))) float v8f;

// ---------------------------------------------------------------------------
// 16x16 tile of T(16x3) * S(16x3)^T via V_WMMA_F32_16X16X4_F32 (K=3 padded to 4)
// A-frag (16x4 f32): lanes 0-15 hold row=lane, K={0,1}; lanes 16-31 K={2,3}.
// B-frag (4x16 f32): mirrored (lane%16 = column).
// D (16x16 f32): lane l holds col=l%16, rows (l<16 ? 0..7 : 8..15) in v[0..7].
// ---------------------------------------------------------------------------
__device__ __forceinline__ v8f wmma_dot_tile(v2f a, v2f b) {
    v8f c = {};
    c = __builtin_amdgcn_wmma_f32_16x16x4_f32(
        /*neg_a=*/false, a, /*neg_b=*/false, b,
        /*c_mod=*/(short)0, c, /*reuse_a=*/false, /*reuse_b=*/false);
    return c;
}

__device__ __forceinline__ v2f load_frag3(const float* p, int hi) {
    // pack (x,y) for K=0,1 half, (z,0) for K=2,3 half
    v2f f;
    f.x = hi ? p[2] : p[0];
    f.y = hi ? 0.0f : p[1];
    return f;
}

// ---------------------------------------------------------------------------
// Pass A: suml[b][n] += sum_m exp(level*d2[n,m]) * remainR[b][m]
// One wave per (batch, 16-row strip, column split). Row-oriented.
// ---------------------------------------------------------------------------
__global__ void k_passA(const float* __restrict__ tpl, const float* __restrict__ src,
                        const float* __restrict__ normT, const float* __restrict__ normS,
                        const float* __restrict__ remainR, float* __restrict__ suml,
                        float level, int N, int M, int nsplit)
{
    const int lane  = threadIdx.x;
    const int ml    = lane & 15;   // tile column / tile row index
    const int hi    = lane >> 4;   // 0: rows 0-7 / K 0,1 ; 1: rows 8-15 / K 2,3
    int blk   = blockIdx.x;
    const int split = blk % nsplit;              blk /= nsplit;
    const int strip = blk % (N / 16);            blk /= (N / 16);
    const int b     = blk;
    const int n0    = strip * 16;

    // A fragment: fixed template rows n0..n0+15
    const float* Trow = tpl + ((size_t)b * N + n0 + ml) * 3;
    const v2f afrag = load_frag3(Trow, hi);

    // |t|^2 for my 8 rows
    const float* pnT = normT + (size_t)b * N + n0 + hi * 8;
    float nT[8];
    #pragma unroll
    for (int v = 0; v < 8; ++v) nT[v] = pnT[v];

    float acc[8];
    #pragma unroll
    for (int v = 0; v < 8; ++v) acc[v] = 0.0f;

    const int   mtiles = (M / nsplit) / 16;
    const int   m0base = split * (M / nsplit);
    const float* S  = src   + (size_t)b * M * 3;
    const float* nS = normS + (size_t)b * M;
    const float* rR = remainR + (size_t)b * M;

    for (int t = 0; t < mtiles; ++t) {
        const int m0 = m0base + t * 16;
        const v2f bfrag = load_frag3(S + (size_t)(m0 + ml) * 3, hi);
        v8f c = wmma_dot_tile(afrag, bfrag);
        const float ns = nS[m0 + ml];
        const float rr = rR[m0 + ml];
        #pragma unroll
        for (int v = 0; v < 8; ++v) {
            float d2 = __builtin_fmaf(-2.0f, c[v], nT[v] + ns);
            acc[v] += __expf(level * d2) * rr;
        }
    }
    // reduce each row-partial across the 16 lanes of each half-wave
    #pragma unroll
    for (int off = 8; off; off >>= 1) {
        #pragma unroll
        for (int v = 0; v < 8; ++v) acc[v] += __shfl_xor(acc[v], off, 32);
    }
    if (ml == 0) {
        float* out = suml + (size_t)b * N + n0 + hi * 8;
        #pragma unroll
        for (int v = 0; v < 8; ++v) atomicAdd(&out[v], acc[v]);
    }
}

// ---------------------------------------------------------------------------
// Pass B: sumr[b][m] += sum_n exp(level*d2[n,m]) * ratioL[b][n]
// One wave per (batch, 16-col strip, row split). Column-oriented.
// ---------------------------------------------------------------------------
__global__ void k_passB(const float* __restrict__ tpl, const float* __restrict__ src,
                        const float* __restrict__ normT, const float* __restrict__ normS,
                        const float* __restrict__ ratioL, float* __restrict__ sumr,
                        float level, int N, int M, int nsplit)
{
    const int lane  = threadIdx.x;
    const int ml    = lane & 15;
    const int hi    = lane >> 4;
    int blk   = blockIdx.x;
    const int split = blk % nsplit;              blk /= nsplit;
    const int strip = blk % (M / 16);            blk /= (M / 16);
    const int b     = blk;
    const int m0    = strip * 16;

    // B fragment: fixed source columns m0..m0+15
    const v2f bfrag = load_frag3(src + ((size_t)b * M + m0 + ml) * 3, hi);
    const float ns  = normS[(size_t)b * M + m0 + ml];

    float sumacc = 0.0f;
    const int   ntiles = (N / nsplit) / 16;
    const int   n0base = split * (N / nsplit);
    const float* T  = tpl    + (size_t)b * N * 3;
    const float* nTb = normT + (size_t)b * N;
    const float* rLb = ratioL + (size_t)b * N;

    for (int t = 0; t < ntiles; ++t) {
        const int n0 = n0base + t * 16;
        const v2f afrag = load_frag3(T + (size_t)(n0 + ml) * 3, hi);
        v8f c = wmma_dot_tile(afrag, bfrag);
        const float* pnT = nTb + n0 + hi * 8;
        const float* prL = rLb + n0 + hi * 8;
        #pragma unroll
        for (int v = 0; v < 8; ++v) {
            float d2 = __builtin_fmaf(-2.0f, c[v], pnT[v] + ns);
            sumacc += __expf(level * d2) * prL[v];
        }
    }
    // each column appears in lane l (rows 0-7) and lane l+16 (rows 8-15)
    sumacc += __shfl_xor(sumacc, 16, 32);
    if (hi == 0) atomicAdd(&sumr[(size_t)b * M + m0 + ml], sumacc);
}

// ---------------------------------------------------------------------------
// Pass C: rowsum[b][n]  += sum_m w0*ratioR[m]
//         costpart[b][n]+= sum_m w0*ratioR[m]*sqrt(max(d2,1e-24))
// ---------------------------------------------------------------------------
__global__ void k_passC(const float* __restrict__ tpl, const float* __restrict__ src,
                        const float* __restrict__ normT, const float* __restrict__ normS,
                        const float* __restrict__ ratioR, float* __restrict__ rowsum,
                        float* __restrict__ costpart,
                        float level, int N, int M, int nsplit)
{
    const int lane  = threadIdx.x;
    const int ml    = lane & 15;
    const int hi    = lane >> 4;
    int blk   = blockIdx.x;
    const int split = blk % nsplit;              blk /= nsplit;
    const int strip = blk % (N / 16);            blk /= (N / 16);
    const int b     = blk;
    const int n0    = strip * 16;

    const v2f afrag = load_frag3(tpl + ((size_t)b * N + n0 + ml) * 3, hi);
    const float* pnT = normT + (size_t)b * N + n0 + hi * 8;
    float nT[8];
    #pragma unroll
    for (int v = 0; v < 8; ++v) nT[v] = pnT[v];

    float accS[8], accC[8];
    #pragma unroll
    for (int v = 0; v < 8; ++v) { accS[v] = 0.0f; accC[v] = 0.0f; }

    const int   mtiles = (M / nsplit) / 16;
    const int   m0base = split * (M / nsplit);
    const float* S  = src   + (size_t)b * M * 3;
    const float* nS = normS + (size_t)b * M;
    const float* rR = ratioR + (size_t)b * M;

    for (int t = 0; t < mtiles; ++t) {
        const int m0 = m0base + t * 16;
        const v2f bfrag = load_frag3(S + (size_t)(m0 + ml) * 3, hi);
        v8f c = wmma_dot_tile(afrag, bfrag);
        const float ns = nS[m0 + ml];
        const float rr = rR[m0 + ml];
        #pragma unroll
        for (int v = 0; v < 8; ++v) {
            float d2 = __builtin_fmaf(-2.0f, c[v], nT[v] + ns);
            float wv = __expf(level * d2) * rr;
            accS[v] += wv;
            accC[v] += wv * sqrtf(fmaxf(d2, 1e-24f));
        }
    }
    #pragma unroll
    for (int off = 8; off; off >>= 1) {
        #pragma unroll
        for (int v = 0; v < 8; ++v) {
            accS[v] += __shfl_xor(accS[v], off, 32);
            accC[v] += __shfl_xor(accC[v], off, 32);
        }
    }
    if (ml == 0) {
        float* os = rowsum   + (size_t)b * N + n0 + hi * 8;
        float* oc = costpart + (size_t)b * N + n0 + hi * 8;
        #pragma unroll
        for (int v = 0; v < 8; ++v) {
            atomicAdd(&os[v], accS[v]);
            atomicAdd(&oc[v], accC[v]);
        }
    }
}

// ---------------------------------------------------------------------------
// Small elementwise kernels
// ---------------------------------------------------------------------------
__global__ void k_zero(float* __restrict__ p, int n) {
    int i = blockIdx.x * blockDim.x + threadIdx.x;
    if (i < n) p[i] = 0.0f;
}

__global__ void k_init(const float* __restrict__ tpl, const float* __restrict__ src,
                       float* __restrict__ remainL, float* __restrict__ remainR,
                       float* __restrict__ normT, float* __restrict__ normS,
                       float* __restrict__ cost,
                       float multiL, float multiR, int BN, int BM, int B)
{
    int i = blockIdx.x * blockDim.x + threadIdx.x;
    if (i < BN) {
        remainL[i] = multiL;
        float x = tpl[3 * (size_t)i], y = tpl[3 * (size_t)i + 1], z = tpl[3 * (size_t)i + 2];
        normT[i] = x * x + y * y + z * z;
    }
    if (i < BM) {
        remainR[i] = multiR;
        float x = src[3 * (size_t)i], y = src[3 * (size_t)i + 1], z = src[3 * (size_t)i + 2];
        normS[i] = x * x + y * y + z * z;
    }
    if (i < B) cost[i] = 0.0f;
}

__global__ void k_ratioL(const float* __restrict__ remainL, const float* __restrict__ suml,
                         float* __restrict__ ratioL, int BN) {
    int i = blockIdx.x * blockDim.x + threadIdx.x;
    if (i < BN) ratioL[i] = remainL[i] / (suml[i] + 1e-9f);
}

__global__ void k_ratioR(float* __restrict__ remainR, const float* __restrict__ sumr,
                         float* __restrict__ ratioR, int BM) {
    int i = blockIdx.x * blockDim.x + threadIdx.x;
    if (i < BM) {
        float s  = sumr[i];
        float r  = remainR[i];
        float rt = r / (s + 1e-9f);
        ratioR[i]  = rt;
        remainR[i] = fmaxf(r - rt * s, 0.0f);   // sum_n w[n,m] == ratioR*sumr_raw
    }
}

__global__ void k_finishL(float* __restrict__ remainL, const float* __restrict__ ratioL,
                          const float* __restrict__ rowsum, const float* __restrict__ costpart,
                          float* __restrict__ cost, int N, int BN)
{
    int i = blockIdx.x * blockDim.x + threadIdx.x;
    float contrib = 0.0f;
    int b = 0;
    if (i < BN) {
        float rl = ratioL[i];
        remainL[i] = fmaxf(remainL[i] - rl * rowsum[i], 0.0f);
        contrib = rl * costpart[i];
        b = i / N;   // whole wave shares one batch (N % 256 == 0)
    }
    #pragma unroll
    for (int off = 16; off; off >>= 1) contrib += __shfl_xor(contrib, off, 32);
    if (((threadIdx.x & 31) == 0) && i < BN) atomicAdd(&cost[b], contrib);
}

__global__ void k_final(const float* __restrict__ cost, float* __restrict__ out, int B, int N) {
    if (blockIdx.x == 0 && threadIdx.x == 0) {
        float s = 0.0f;
        for (int b = 0; b < B; ++b) s += cost[b];
        out[0] = s / ((float)B * (float)N);
    }
}

// ---------------------------------------------------------------------------
extern "C" void kernel_launch(void* const* d_in, const int* in_sizes, int n_in,
                              void* d_out, int out_size, void* d_ws, size_t ws_size,
                              hipStream_t stream)
{
    const float* tpl = (const float*)d_in[0];
    const float* src = (const float*)d_in[1];
    float*       out = (float*)d_out;

    const int B = 4;
    const int N = in_sizes[0] / (B * 3);   // 2048
    const int M = in_sizes[1] / (B * 3);   // 2048
    const int BN = B * N, BM = B * M;

    float* ws       = (float*)d_ws;
    float* suml     = ws;            ws += BN;
    float* ratioL   = ws;            ws += BN;
    float* sumr     = ws;            ws += BM;
    float* ratioR   = ws;            ws += BM;
    float* remainL  = ws;            ws += BN;
    float* remainR  = ws;            ws += BM;
    float* rowsum   = ws;            ws += BN;
    float* costpart = ws;            ws += BN;
    float* normT    = ws;            ws += BN;
    float* normS    = ws;            ws += BM;
    float* cost     = ws;            ws += B;

    float multiL, multiR;
    if (N >= M) { multiL = 1.0f; multiR = (float)(N / M); }
    else        { multiL = (float)(M / N); multiR = 1.0f; }

    const int initN = (BN > BM ? BN : BM);
    k_init<<<(initN + 255) / 256, 256, 0, stream>>>(tpl, src, remainL, remainR,
                                                    normT, normS, cost,
                                                    multiL, multiR, BN, BM, B);

    const int NSPLIT = 8;                       // waves per strip for occupancy
    const int gA = B * (N / 16) * NSPLIT;       // row-strip waves
    const int gB = B * (M / 16) * NSPLIT;       // col-strip waves
    const dim3 wblk(32);

    for (int j = 0; j < 10; ++j) {
        // levels: -4^(7-j) for j=0..8, then 0
        const float level = (j == 9) ? 0.0f : -ldexpf(1.0f, 2 * (7 - j));

        k_zero<<<(BN + 255) / 256, 256, 0, stream>>>(suml, BN);
        k_passA<<<gA, wblk, 0, stream>>>(tpl, src, normT, normS, remainR, suml,
                                         level, N, M, NSPLIT);
        k_ratioL<<<(BN + 255) / 256, 256, 0, stream>>>(remainL, suml, ratioL, BN);

        k_zero<<<(BM + 255) / 256, 256, 0, stream>>>(sumr, BM);
        k_passB<<<gB, wblk, 0, stream>>>(tpl, src, normT, normS, ratioL, sumr,
                                         level, N, M, NSPLIT);
        k_ratioR<<<(BM + 255) / 256, 256, 0, stream>>>(remainR, sumr, ratioR, BM);

        k_zero<<<(BN + 255) / 256, 256, 0, stream>>>(rowsum, BN);
        k_zero<<<(BN + 255) / 256, 256, 0, stream>>>(costpart, BN);
        k_passC<<<gA, wblk, 0, stream>>>(tpl, src, normT, normS, ratioR,
                                         rowsum, costpart, level, N, M, NSPLIT);
        k_finishL<<<(BN + 255) / 256, 256, 0, stream>>>(remainL, ratioL, rowsum,
                                                        costpart, cost, N, BN);
    }
    k_final<<<1, 1, 0, stream>>>(cost, out, B, N);
}